// NonMaxSuppression_71846212927714
// MI455X (gfx1250) — compile-verified
//
#include <hip/hip_runtime.h>

#define NCLS   80
#define NBOX   2048
#define NB     4
#define CONF_T 0.05f
#define IOU_T  0.5f
#define MAXDET 100

typedef unsigned int   u32;
typedef unsigned short u16;
typedef u32 u32x4 __attribute__((ext_vector_type(4)));
typedef u32 u32x8 __attribute__((ext_vector_type(8)));

// Single shared struct so `pred` sits at LDS offset 0 — the TDM descriptor's
// lds_addr targets offset 0 directly.  Total 57,608 B (< 64 KB, well under the
// 320 KB/WGP CDNA5 budget).
struct SmemB {
  float pred[NBOX * 6];   // staged by tensor_load_to_lds (48 KB)
  u16   candIdx[NBOX];    // compacted candidate -> original box index
  u16   sortedId[NBOX];   // rank -> candidate slot
  u32   supp[NBOX / 32];  // suppression bitmask over sorted positions
  int   candCnt;
  int   keptCnt;
};

__global__ __launch_bounds__(256) void nms_per_class(
    const float* __restrict__ pred,
    float* __restrict__ keptScore,   // [NB][NCLS][MAXDET]
    int*   __restrict__ keptIdx,     // [NB][NCLS][MAXDET]
    int*   __restrict__ keptNum) {   // [NB][NCLS]
  __shared__ SmemB sm;
  const int c   = blockIdx.x;
  const int b   = blockIdx.y;
  const int tid = threadIdx.x;

  // ---- Stage this image's predictions (2048x6 f32 = 48 KB) into LDS via TDM.
  // 1-D tensor: data_size=4B (code 2), tensor_dim0 = tile_dim0 = 12288,
  // tensor_dim1 = 1, stride0 = 12288, lds_addr = 0, D# type = 2 ("image").
  if (tid < 32) {  // one wave issues the DMA; EXEC is ignored by TDM
    unsigned long long ga = (unsigned long long)(pred + (size_t)b * NBOX * 6);
    u32x4 g0;
    g0.x = 1u;                                              // count=1, user D#
    g0.y = 0u;                                              // lds_addr = 0
    g0.z = (u32)ga;                                         // global_addr lo32
    g0.w = ((u32)(ga >> 32) & 0x01FFFFFFu) | 0x80000000u;   // addr hi + type=2
    u32x8 g1;
    g1.s0 = 0x00020000u;                  // wg_mask=0, data_size=4B
    g1.s1 = (u32)(NBOX * 6) << 16;        // tensor_dim0[15:0] in [31:16]
    g1.s2 = 0x00010000u;                  // tensor_dim0 hi=0, tensor_dim1=1
    g1.s3 = (u32)(NBOX * 6) << 16;        // tile_dim0 = 12288
    g1.s4 = 0u;                           // tile_dim1=0, tile_dim2=0
    g1.s5 = (u32)(NBOX * 6);              // tensor_dim0_stride lo32
    g1.s6 = 0u;
    g1.s7 = 0u;
    asm volatile("tensor_load_to_lds %0, %1" :: "s"(g0), "s"(g1) : "memory");
    __builtin_amdgcn_s_wait_tensorcnt(0);
  }
  if (tid == 0) { sm.candCnt = 0; sm.keptCnt = 0; }
  for (int i = tid; i < NBOX / 32; i += 256) sm.supp[i] = 0u;
  __syncthreads();

  // ---- Gather candidates of this class above the confidence threshold.
  for (int n = tid; n < NBOX; n += 256) {
    float s  = sm.pred[n * 6 + 5];
    int   cl = (int)sm.pred[n * 6 + 4];
    if (cl == c && s > CONF_T) {
      int slot = atomicAdd(&sm.candCnt, 1);   // ds_add_rtn_u32
      sm.candIdx[slot] = (u16)n;
    }
  }
  __syncthreads();
  const int M = sm.candCnt;   // expected ~26, worst 2048

  // ---- Enumeration sort: rank by (score desc, original index asc) — matches
  // the stability of jnp.argsort(-scores).
  for (int t = tid; t < M; t += 256) {
    int   pi = sm.candIdx[t];
    float si = sm.pred[pi * 6 + 5];
    int   r  = 0;
    for (int j = 0; j < M; ++j) {
      int   pj = sm.candIdx[j];
      float sj = sm.pred[pj * 6 + 5];
      r += (sj > si) || (sj == si && pj < pi);
    }
    sm.sortedId[r] = (u16)t;
  }
  __syncthreads();

  // ---- Greedy suppression over sorted order.  A kept box beyond the 100/class
  // cap still suppresses (reference applies the cumsum cap after the scan).
  const int gb = (b * NCLS + c) * MAXDET;
  for (int i = 0; i < M; ++i) {
    __syncthreads();
    const bool kept_i = (((sm.supp[i >> 5] >> (i & 31)) & 1u) == 0u);
    if (kept_i) {
      const int   ci = sm.candIdx[sm.sortedId[i]];
      const float x1 = sm.pred[ci * 6 + 0], y1 = sm.pred[ci * 6 + 1];
      const float x2 = sm.pred[ci * 6 + 2], y2 = sm.pred[ci * 6 + 3];
      if (tid == 0) {
        int k = sm.keptCnt;
        if (k < MAXDET) {
          keptScore[gb + k] = sm.pred[ci * 6 + 5];
          keptIdx[gb + k]   = ci;
          sm.keptCnt        = k + 1;
        }
      }
      const float ai = (x2 - x1) * (y2 - y1);
      for (int j = i + 1 + tid; j < M; j += 256) {
        if ((sm.supp[j >> 5] >> (j & 31)) & 1u) continue;
        const int   cj  = sm.candIdx[sm.sortedId[j]];
        const float bx1 = sm.pred[cj * 6 + 0], by1 = sm.pred[cj * 6 + 1];
        const float bx2 = sm.pred[cj * 6 + 2], by2 = sm.pred[cj * 6 + 3];
        const float iw    = fminf(x2, bx2) - fmaxf(x1, bx1);
        const float ih    = fminf(y2, by2) - fmaxf(y1, by1);
        const float inter = fmaxf(iw, 0.0f) * fmaxf(ih, 0.0f);
        const float aj    = (bx2 - bx1) * (by2 - by1);
        const float un    = ai + aj - inter;
        const float iou   = (un > 0.0f) ? inter / un : 0.0f;
        if (iou > IOU_T) atomicOr(&sm.supp[j >> 5], 1u << (j & 31));  // ds_or_b32
      }
    }
  }
  __syncthreads();
  if (tid == 0) keptNum[b * NCLS + c] = sm.keptCnt;
}

// ---- Kernel 2: per-image global top-100 over all kept detections. ----------
struct SmemC {
  float sc[NCLS * MAXDET];    // 32,000 B
  u32   info[NCLS * MAXDET];  // 32,000 B : origIdx<<16 | cls<<8 | slot
  int   tot;
};

__global__ __launch_bounds__(256) void topk_write(
    const float* __restrict__ pred,
    const float* __restrict__ keptScore,
    const int*   __restrict__ keptIdx,
    const int*   __restrict__ keptNum,
    float*       __restrict__ out,       // [NB][MAXDET][6]
    int*         __restrict__ validOut) {
  __shared__ SmemC sm;
  const int b   = blockIdx.x;
  const int tid = threadIdx.x;
  if (tid == 0) sm.tot = 0;
  __syncthreads();

  // Collect kept entries.  Tie-break key (cls<<8 | slot) reproduces top_k's
  // lower-flat-index-wins ordering: flat index was class-major, and per-class
  // kept slots are emitted in ascending sorted position.
  for (int c = tid; c < NCLS; c += 256) {
    const int n    = keptNum[b * NCLS + c];
    const int base = atomicAdd(&sm.tot, n);
    const int gb   = (b * NCLS + c) * MAXDET;
    for (int k = 0; k < n; ++k) {
      sm.sc[base + k]   = keptScore[gb + k];
      sm.info[base + k] = ((u32)keptIdx[gb + k] << 16) | ((u32)c << 8) | (u32)k;
    }
  }
  __syncthreads();

  // Zero the whole output block first (harness poisons d_out).
  for (int i = tid; i < MAXDET * 6; i += 256) out[(size_t)b * MAXDET * 6 + i] = 0.0f;
  const int T = sm.tot;
  if (tid == 0) validOut[b] = (T < MAXDET) ? T : MAXDET;
  __syncthreads();

  // Rank each entry; ranks are unique, so top-100 writes are conflict-free.
  for (int e = tid; e < T; e += 256) {
    const float se = sm.sc[e];
    const u32   pe = sm.info[e] & 0xFFFFu;
    int r = 0;
    for (int f = 0; f < T; ++f) {
      const float sf = sm.sc[f];
      r += (sf > se) || (sf == se && (sm.info[f] & 0xFFFFu) < pe);
    }
    if (r < MAXDET) {
      const u32 inf  = sm.info[e];
      const int orig = (int)(inf >> 16);
      const int cls  = (int)((inf >> 8) & 0xFFu);
      const float* p = pred + ((size_t)b * NBOX + orig) * 6;
      __builtin_prefetch(p, 0, 1);   // global_prefetch_b8
      float* o = out + (size_t)b * MAXDET * 6 + (size_t)r * 6;
      o[0] = p[0]; o[1] = p[1]; o[2] = p[2]; o[3] = p[3];
      o[4] = (float)cls; o[5] = se;
    }
  }
}

extern "C" void kernel_launch(void* const* d_in, const int* in_sizes, int n_in,
                              void* d_out, int out_size, void* d_ws, size_t ws_size,
                              hipStream_t stream) {
  (void)in_sizes; (void)n_in; (void)out_size; (void)ws_size;
  const float* pred = (const float*)d_in[0];
  float* out = (float*)d_out;
  int*   validOut = (int*)((float*)d_out + (size_t)NB * MAXDET * 6);

  char*  ws        = (char*)d_ws;
  float* keptScore = (float*)ws;                                        // 128,000 B
  int*   keptIdx   = (int*)(ws + (size_t)NB * NCLS * MAXDET * 4);       // 128,000 B
  int*   keptNum   = (int*)(ws + (size_t)2 * NB * NCLS * MAXDET * 4);   //   1,280 B

  nms_per_class<<<dim3(NCLS, NB), 256, 0, stream>>>(pred, keptScore, keptIdx, keptNum);
  topk_write<<<NB, 256, 0, stream>>>(pred, keptScore, keptIdx, keptNum, out, validOut);
}